// GATLayer_35476429865592
// MI455X (gfx1250) — compile-verified
//
#include <hip/hip_runtime.h>

#define N_NODES 50000
#define N_EDGES 1600000
#define IN_DIM 256
#define OUT_DIM 128
#define N_HEADS 4
#define NEG_SLOPE 0.2f

typedef float v2f __attribute__((ext_vector_type(2)));
typedef float v8f __attribute__((ext_vector_type(8)));

// Workspace layout (floats):
//   [0,            N*8)   s    : s[i*8+h] = s_src[i][h], s[i*8+4+h] = s_dst[i][h]
//   [N*8,          N*12)  denom: denom[i*4+h]
//   [N*12,         N*140) fch  : fch[i*128+c]

__global__ __launch_bounds__(256) void zero_kernel(float* __restrict__ denom,
                                                   float* __restrict__ out) {
    int i = blockIdx.x * blockDim.x + threadIdx.x;
    if (i < N_NODES * N_HEADS) denom[i] = 0.f;
    if (i < N_NODES * OUT_DIM) out[i] = 0.f;
}

// s = x @ [w_src.T | w_dst.T]  -> N x 8, one wave per 16-row tile.
// B cols 0..3 = src heads, 4..7 = dst heads, 8..15 = zero padding.
// Padding lanes load a valid dummy address and zero the value with a mask
// multiply -- keeps the B fetch a single unconditional global_load_b64
// (no EXEC save/restore in the inner loop; WMMA needs EXEC all-1s anyway).
__global__ __launch_bounds__(32) void attn_score_wmma(const float* __restrict__ x,
                                                      const float* __restrict__ attn_w,
                                                      float* __restrict__ s) {
    const int lane    = threadIdx.x;        // 0..31
    const int rowBase = blockIdx.x * 16;
    const int n       = lane & 15;          // matrix column (B/D) / A row
    const int kh      = (lane >> 4) * 2;    // K sub-offset: 0 or 2

    const bool  bactive = (n < 8);
    const float bmask   = bactive ? 1.f : 0.f;
    const int   nc      = bactive ? n : 0;  // clamp padding lanes to col 0
    const int   boff    = (nc & 3) * (2 * IN_DIM) + ((nc & 4) ? IN_DIM : 0);

    v8f acc = {};
    const float* arow = x + (size_t)(rowBase + n) * IN_DIM;
    const float* brow = attn_w + boff;
#pragma unroll 4
    for (int k0 = 0; k0 < IN_DIM; k0 += 4) {
        const int k = k0 + kh;
        v2f a = *(const v2f*)(arow + k);
        v2f b = *(const v2f*)(brow + k);
        b.x *= bmask;
        b.y *= bmask;
        acc = __builtin_amdgcn_wmma_f32_16x16x4_f32(false, a, false, b,
                                                    (short)0, acc, false, false);
    }
    if (bactive) {
#pragma unroll
        for (int v = 0; v < 8; ++v) {
            const int row = rowBase + (lane >> 4) * 8 + v;
            s[(size_t)row * 8 + n] = acc[v];
        }
    }
}

// fch = x @ fc_w.T + fc_b.  Block = 8 waves; wave w owns a 16-col tile.
__global__ __launch_bounds__(256) void fc_wmma(const float* __restrict__ x,
                                               const float* __restrict__ fc_w,
                                               const float* __restrict__ fc_b,
                                               float* __restrict__ fch) {
    const int lane    = threadIdx.x & 31;
    const int wave    = threadIdx.x >> 5;   // 0..7
    const int rowBase = blockIdx.x * 16;
    const int colBase = wave * 16;
    const int n       = lane & 15;
    const int kh      = (lane >> 4) * 2;

    const float* arow = x    + (size_t)(rowBase + n) * IN_DIM;
    const float* brow = fc_w + (size_t)(colBase + n) * IN_DIM;   // B[k][n] = fc_w[n][k]

    v8f acc = {};
#pragma unroll 4
    for (int k0 = 0; k0 < IN_DIM; k0 += 4) {
        v2f a = *(const v2f*)(arow + k0 + kh);
        v2f b = *(const v2f*)(brow + k0 + kh);
        acc = __builtin_amdgcn_wmma_f32_16x16x4_f32(false, a, false, b,
                                                    (short)0, acc, false, false);
    }
    const float bias = fc_b[colBase + n];
#pragma unroll
    for (int v = 0; v < 8; ++v) {
        const int row = rowBase + (lane >> 4) * 8 + v;
        fch[(size_t)row * OUT_DIM + colBase + n] = acc[v] + bias;
    }
}

// Pass 1 over edges: denom[src] += leaky(e). One thread per edge, 4 f32 atomics.
__global__ __launch_bounds__(256) void edge_denom(const long long* __restrict__ ei,
                                                  const float* __restrict__ s,
                                                  const float* __restrict__ attn_b,
                                                  float* __restrict__ denom) {
    int e = blockIdx.x * blockDim.x + threadIdx.x;
    if (e >= N_EDGES) return;
    const int src = (int)ei[e];
    const int dst = (int)ei[N_EDGES + e];
#pragma unroll
    for (int h = 0; h < N_HEADS; ++h) {
        float ev = s[(size_t)src * 8 + h] + s[(size_t)dst * 8 + 4 + h] + attn_b[h];
        ev = (ev >= 0.f) ? ev : NEG_SLOPE * ev;
        unsafeAtomicAdd(&denom[(size_t)src * N_HEADS + h], ev);
    }
}

// Pass 2 over edges: out[src] += coeff * fch[dst]. One wave per edge; each lane
// owns a float4 slice of the 128-wide feature. Recomputes e (cheaper than
// storing E*4 floats). All gathers hit the 192MB L2 (fch/out/s/denom resident).
__global__ __launch_bounds__(256) void edge_aggregate(const long long* __restrict__ ei,
                                                      const float* __restrict__ s,
                                                      const float* __restrict__ attn_b,
                                                      const float* __restrict__ denom,
                                                      const float* __restrict__ fch,
                                                      float* __restrict__ out) {
    const int gid  = blockIdx.x * blockDim.x + threadIdx.x;
    const int e    = gid >> 5;
    const int lane = gid & 31;
    if (e >= N_EDGES) return;
    const int src = (int)ei[e];
    const int dst = (int)ei[N_EDGES + e];

    float c = 0.f;
#pragma unroll
    for (int h = 0; h < N_HEADS; ++h) {
        float ev = s[(size_t)src * 8 + h] + s[(size_t)dst * 8 + 4 + h] + attn_b[h];
        ev = (ev >= 0.f) ? ev : NEG_SLOPE * ev;
        c += ev / denom[(size_t)src * N_HEADS + h];
    }
    c *= (1.0f / N_HEADS);

    const float4 v = *(const float4*)(fch + (size_t)dst * OUT_DIM + lane * 4);
    float* o = out + (size_t)src * OUT_DIM + lane * 4;
    unsafeAtomicAdd(o + 0, c * v.x);
    unsafeAtomicAdd(o + 1, c * v.y);
    unsafeAtomicAdd(o + 2, c * v.z);
    unsafeAtomicAdd(o + 3, c * v.w);
}

__global__ __launch_bounds__(256) void relu_kernel(float* __restrict__ out) {
    int i = blockIdx.x * blockDim.x + threadIdx.x;
    if (i < N_NODES * OUT_DIM) out[i] = fmaxf(out[i], 0.f);
}

extern "C" void kernel_launch(void* const* d_in, const int* in_sizes, int n_in,
                              void* d_out, int out_size, void* d_ws, size_t ws_size,
                              hipStream_t stream) {
    const float*     x      = (const float*)d_in[0];
    const long long* ei     = (const long long*)d_in[1];   // int64 edge_index, shape (2,E)
    const float*     attn_w = (const float*)d_in[2];
    const float*     attn_b = (const float*)d_in[3];
    const float*     fc_w   = (const float*)d_in[4];
    const float*     fc_b   = (const float*)d_in[5];
    float* out = (float*)d_out;

    float* ws    = (float*)d_ws;
    float* s     = ws;                              // N*8
    float* denom = ws + (size_t)N_NODES * 8;        // N*4
    float* fch   = ws + (size_t)N_NODES * 12;       // N*128

    zero_kernel<<<(N_NODES * OUT_DIM + 255) / 256, 256, 0, stream>>>(denom, out);
    attn_score_wmma<<<N_NODES / 16, 32, 0, stream>>>(x, attn_w, s);
    fc_wmma<<<N_NODES / 16, 256, 0, stream>>>(x, fc_w, fc_b, fch);
    edge_denom<<<(N_EDGES + 255) / 256, 256, 0, stream>>>(ei, s, attn_b, denom);
    edge_aggregate<<<(N_EDGES * 32 + 255) / 256, 256, 0, stream>>>(ei, s, attn_b, denom, fch, out);
    relu_kernel<<<(N_NODES * OUT_DIM + 255) / 256, 256, 0, stream>>>(out);
}